// EdgeBlock_45595372814909
// MI455X (gfx1250) — compile-verified
//
#include <hip/hip_runtime.h>

typedef __attribute__((ext_vector_type(2))) float v2f;
typedef __attribute__((ext_vector_type(4))) float v4f;
typedef __attribute__((ext_vector_type(8))) float v8f;

#define B_ 4
#define L_ 1024
#define N_ 16
#define D_ 64
#define W_ 16
#define V_ 33
#define NEG_SLOPE_ 5.0f
#define MASK_FILL_ (-1000000000000.0f)

// ---------------------------------------------------------------------------
// qc[b,l,n,h] = sum_d hidden[b,l,n,d] * cross[n,d,h]
// One wave per 16(M=l) x 16(N=h) tile; K=64 -> 16 chained WMMA f32 16x16x4.
// ---------------------------------------------------------------------------
__global__ __launch_bounds__(32) void qc_kernel(const float* __restrict__ hidden,
                                                const float* __restrict__ cross,
                                                float* __restrict__ qc) {
    int id = blockIdx.x;
    int ht = id & 3;  id >>= 2;     // D/16 = 4 h-tiles
    int lt = id & 63; id >>= 6;     // L/16 = 64 l-tiles
    int n  = id & 15; id >>= 4;     // N = 16
    int b  = id;
    int lane = threadIdx.x;
    int m    = lane & 15;
    int ksel = lane >> 4;           // 0: K={0,1}, 1: K={2,3} within 4-wide K step
    int l0 = lt * 16;
    int h0 = ht * 16;

    // A: hidden rows (M=l0+m), elements K = k0 + 2*ksel + {0,1}
    const float* Ab = hidden + (((size_t)(b * L_ + l0 + m) * N_ + n) * D_) + 2 * ksel;
    // B: cross[n, k, h0+col], col = m, rows K = k0 + 2*ksel + {0,1}
    const float* Bb = cross + ((size_t)n * D_ + 2 * ksel) * D_ + h0 + m;

    v8f c = {};
    #pragma unroll
    for (int k0 = 0; k0 < D_; k0 += 4) {
        v2f a, bb;
        a.x  = Ab[k0];
        a.y  = Ab[k0 + 1];
        bb.x = Bb[(size_t)k0 * D_];
        bb.y = Bb[(size_t)(k0 + 1) * D_];
        c = __builtin_amdgcn_wmma_f32_16x16x4_f32(false, a, false, bb,
                                                  (short)0, c, false, false);
    }
    // D layout: VGPR r -> row (r + 8*ksel), col = m
    #pragma unroll
    for (int r = 0; r < 8; ++r) {
        int row = r + 8 * ksel;
        qc[(((size_t)(b * L_ + l0 + row) * N_ + n) * D_) + h0 + m] = c[r];
    }
}

// ---------------------------------------------------------------------------
// upon_s[b,n,l] = dot(hidden[b,l,n,:], upon[n,:]); same for down_s.
// ---------------------------------------------------------------------------
__global__ __launch_bounds__(256) void scores_kernel(const float* __restrict__ hidden,
                                                     const float* __restrict__ upon,
                                                     const float* __restrict__ down,
                                                     float* __restrict__ upon_s,
                                                     float* __restrict__ down_s) {
    int t = blockIdx.x * blockDim.x + threadIdx.x;   // over B*N*L, l fastest
    if (t >= B_ * N_ * L_) return;
    int l = t % L_;
    int n = (t / L_) % N_;
    int b = t / (L_ * N_);
    const float* h = hidden + (((size_t)(b * L_ + l) * N_ + n) * D_);
    const float* u = upon + n * D_;
    const float* d = down + n * D_;
    float su = 0.f, sd = 0.f;
    #pragma unroll 8
    for (int k = 0; k < D_; ++k) {
        float x = h[k];
        su += x * u[k];
        sd += x * d[k];
    }
    upon_s[t] = su;
    down_s[t] = sd;
}

// ---------------------------------------------------------------------------
// Banded product + epilogue. One wave per (b, n, 16-wide l-tile).
// P[i,jj] = sum_h qc[b,l0+i,n,h] * hidden[b,(l0+jj)%L,n,h], jj in [0,32)
// (off_u in [0,16] => idx_u for the tile spans exactly [l0, l0+31] mod L).
// Then window_score[b,n,l0+i,v] = masked leaky_relu(concat + P[i, i+off_u(v)]).
// ---------------------------------------------------------------------------
__global__ __launch_bounds__(32) void win_kernel(const float* __restrict__ hidden,
                                                 const float* __restrict__ qc,
                                                 const float* __restrict__ upon_s,
                                                 const float* __restrict__ down_s,
                                                 const unsigned char* __restrict__ mask,
                                                 float* __restrict__ wscore) {
    int id = blockIdx.x;
    int lt = id & 63; id >>= 6;     // L/16
    int n  = id & 15; id >>= 4;     // N
    int b  = id;
    int l0 = lt * 16;
    int lane = threadIdx.x;
    int m    = lane & 15;
    int ksel = lane >> 4;

    __shared__ float P[16][33];     // 16 rows x 32 cols (+1 pad)

    const float* Ab = qc + (((size_t)(b * L_ + l0 + m) * N_ + n) * D_) + 2 * ksel;
    int j0 = l0 + m;        if (j0 >= L_) j0 -= L_;
    int j1 = l0 + 16 + m;   if (j1 >= L_) j1 -= L_;
    const float* B0 = hidden + (((size_t)(b * L_ + j0) * N_ + n) * D_) + 2 * ksel;
    const float* B1 = hidden + (((size_t)(b * L_ + j1) * N_ + n) * D_) + 2 * ksel;

    v8f c0 = {}, c1 = {};
    #pragma unroll
    for (int k0 = 0; k0 < D_; k0 += 4) {
        v2f a, b0, b1;
        a.x  = Ab[k0];      a.y  = Ab[k0 + 1];
        b0.x = B0[k0];      b0.y = B0[k0 + 1];
        b1.x = B1[k0];      b1.y = B1[k0 + 1];
        c0 = __builtin_amdgcn_wmma_f32_16x16x4_f32(false, a, false, b0,
                                                   (short)0, c0, false, false);
        c1 = __builtin_amdgcn_wmma_f32_16x16x4_f32(false, a, false, b1,
                                                   (short)0, c1, false, false);
    }
    #pragma unroll
    for (int r = 0; r < 8; ++r) {
        int row = r + 8 * ksel;
        P[row][m]      = c0[r];
        P[row][16 + m] = c1[r];
    }
    __syncthreads();

    const float* us = upon_s + ((size_t)b * N_ + n) * L_;
    const float* ds = down_s + ((size_t)b * N_ + n) * L_;
    const unsigned char* mk = mask + (size_t)b * L_;
    float* out = wscore + (((size_t)(b * N_ + n) * L_) + l0) * V_;

    for (int t = lane; t < 16 * V_; t += 32) {
        int i = t / V_;
        int v = t - i * V_;
        int l = l0 + i;
        int ou = (v <= W_) ? 0 : (v - W_);      // off_u
        int od = (v <  W_) ? (v - W_) : 0;      // off_d
        int iu = l + ou;
        int idd = l + od;
        bool valid = (idd >= 0) && (iu < L_);
        int ju = (iu >= L_) ? (iu - L_) : iu;   // idx_u % L
        int jd = (idd < 0) ? (idd + L_) : idd;  // idx_d % L
        float concat = us[ju] + ds[jd];
        float s = concat + P[i][i + ou];
        s = (s >= 0.f) ? s : NEG_SLOPE_ * s;    // leaky_relu, slope 5
        bool mm = (mk[ju] != 0) && valid;
        float r = mm ? s : MASK_FILL_;
        // Streaming output: non-temporal so it doesn't occupy L2.
        __builtin_nontemporal_store(r, &out[(size_t)i * V_ + v]);
    }
}

// ---------------------------------------------------------------------------
// value_h[b,l,v,:,:] = hidden[b, idx_u(l,v), :, :]  — 553 MB streaming copy.
// One 256-thread block per (b,l,v): 1024 contiguous floats, 1 v4f/thread.
// Source (16 MB) is reused 33x and fits in the 192 MB L2 -> regular loads;
// destination is written once -> non-temporal stores keep L2 for the source.
// ---------------------------------------------------------------------------
__global__ __launch_bounds__(256) void gather_kernel(const float* __restrict__ hidden,
                                                     float* __restrict__ value_h) {
    int id = blockIdx.x;                         // over B*L*V, v fastest
    int v = id % V_;
    int l = (id / V_) % L_;
    int b = id / (V_ * L_);
    int ou = (v <= W_) ? 0 : (v - W_);
    int j = l + ou; if (j >= L_) j -= L_;
    const v4f* __restrict__ src =
        (const v4f*)(hidden + ((size_t)(b * L_ + j)) * (N_ * D_));
    v4f* __restrict__ dst = (v4f*)(value_h + ((size_t)id) * (N_ * D_));
    v4f x = src[threadIdx.x];
    __builtin_nontemporal_store(x, &dst[threadIdx.x]);
}

extern "C" void kernel_launch(void* const* d_in, const int* in_sizes, int n_in,
                              void* d_out, int out_size, void* d_ws, size_t ws_size,
                              hipStream_t stream) {
    (void)in_sizes; (void)n_in; (void)out_size; (void)ws_size;
    const float*         hidden = (const float*)d_in[0];
    const unsigned char* mask   = (const unsigned char*)d_in[1];   // jnp bool -> 1 byte
    const float*         upon   = (const float*)d_in[2];
    const float*         down   = (const float*)d_in[3];
    const float*         cross  = (const float*)d_in[4];
    // d_in[5] = window_size (scalar); W_=16 fixed at compile time (shapes depend on it)

    float* ws     = (float*)d_ws;
    float* qc     = ws;                                        // B*L*N*D   (16 MB)
    float* upon_s = qc + (size_t)B_ * L_ * N_ * D_;            // B*N*L
    float* down_s = upon_s + (size_t)B_ * N_ * L_;             // B*N*L

    float* wscore  = (float*)d_out;                            // B*N*L*V
    float* value_h = wscore + (size_t)B_ * N_ * L_ * V_;       // B*L*V*N*D

    qc_kernel<<<B_ * N_ * (L_ / 16) * (D_ / 16), 32, 0, stream>>>(hidden, cross, qc);

    int nsc = B_ * N_ * L_;
    scores_kernel<<<(nsc + 255) / 256, 256, 0, stream>>>(hidden, upon, down, upon_s, down_s);

    win_kernel<<<B_ * N_ * (L_ / 16), 32, 0, stream>>>(hidden, qc, upon_s, down_s, mask, wscore);

    gather_kernel<<<B_ * L_ * V_, 256, 0, stream>>>(hidden, value_h);
}